// TOGL_2156073583135
// MI455X (gfx1250) — compile-verified
//
#include <hip/hip_runtime.h>
#include <hip/hip_bf16.h>
#include <math.h>

#define N_NODES   10000
#define N_FEAT    128
#define HIDDEN    64
#define N_FILT    4
#define N_EDGES   80000      // directed (undirected doubled by reference)
#define N_TILES   (N_NODES / 16)          // 625 row tiles of 16
#define WAVES_PB  4                       // 4 waves (tiles) per block
#define SORT_N    16384      // next pow2 >= N_NODES
#define PERS_LDS_BYTES ((SORT_N * 2 + 3 * N_NODES) * 4)   // 251072 B < 320 KB WGP LDS

typedef __attribute__((ext_vector_type(2))) float v2f;
typedef __attribute__((ext_vector_type(8))) float v8f;

// LDS byte offset of a __shared__ object (generic -> AS(3) -> int)
__device__ __forceinline__ unsigned lds_offset(const void* p)
{
    return (unsigned)(unsigned long long)
        (__attribute__((address_space(3))) const void*)p;
}

// ---------------------------------------------------------------------------
// Kernel 1: F = relu(X@W1 + b1) @ W2 + b2  via V_WMMA_F32_16X16X4_F32.
// 4 waves per block, one 16-row tile per wave. W1 (32 KB) is staged once per
// block into LDS with gfx1250 per-lane async global->LDS B32 transfers
// (ASYNCcnt), *swizzled* into k-pair-interleaved form so each WMMA B operand
// is one naturally aligned ds_load_b64 straight into an even VGPR pair.
// fp32 WMMA keeps rankings bit-faithful for the downstream argsort.
// ---------------------------------------------------------------------------
__global__ __launch_bounds__(WAVES_PB * 32)
void togl_gemm_kernel(const float* __restrict__ X,  const float* __restrict__ W1,
                      const float* __restrict__ b1, const float* __restrict__ W2,
                      const float* __restrict__ b2, float* __restrict__ F)
{
    // w1s[(k>>1)*HIDDEN + n][2] = { W1[k][n], W1[k+1][n] }
    __shared__ __align__(16) float w1s[N_FEAT * HIDDEN];   // 32 KB staged W1 (pair-interleaved)
    __shared__ float hid[WAVES_PB][16 * HIDDEN];           // 16 KB relu'd hidden tiles

    const int tid  = threadIdx.x;                          // 0..127
    const int lane = tid & 31;
    const int wave = tid >> 5;
    const int tile = blockIdx.x * WAVES_PB + wave;         // 0..627 (625 valid)
    const bool valid = (tile < N_TILES);
    const int row0 = tile * 16;
    const int m15  = lane & 15;
    const int koff = (lane >> 4) * 2;    // lanes 0-15 -> K 0,1 ; lanes 16-31 -> K 2,3

    // ---- async stage + swizzle W1 into LDS: 8192 per-lane B32 transfers ----
    {
        const unsigned base = lds_offset(w1s);
#pragma unroll 8
        for (int j = 0; j < N_FEAT * HIDDEN / (WAVES_PB * 32); ++j) {   // 64 per thread
            const int g = tid + 128 * j;                   // flat W1 element
            const int k = g >> 6;                          // row (0..127)
            const int n = g & 63;                          // col (0..63)
            const unsigned loff = base + (((unsigned)(k >> 1) * HIDDEN + n) * 2u
                                          + (unsigned)(k & 1)) * 4u;
            const float* gp = W1 + g;
            asm volatile("global_load_async_to_lds_b32 %0, %1, off"
                         :: "v"(loff), "v"(gp) : "memory");
        }
        asm volatile("s_wait_asynccnt 0x0" ::: "memory");
    }
    __syncthreads();

    // ---- WMMA k-loop (A from global, B as single ds_load_b64 fragments) ----
    v8f acc0 = {}, acc1 = {}, acc2 = {}, acc3 = {};
    const int rowA = valid ? (row0 + m15) : (N_NODES - 1); // clamp ragged tail
    const float* xrow = X + (size_t)rowA * N_FEAT;

    for (int k = 0; k < N_FEAT; k += 4) {
        // A tile 16x4 (f32): VGPR0 = K=koff, VGPR1 = K=koff+1 per ISA layout
        v2f a;
        a.x = xrow[k + koff];
        a.y = xrow[k + koff + 1];
        // B tile 4x16: (K,K+1) pair contiguous in LDS -> one b64 per fragment
        const v2f* wp = (const v2f*)(w1s + (size_t)((k + koff) >> 1) * HIDDEN * 2);
        const v2f vb0 = wp[ 0 + m15];
        const v2f vb1 = wp[16 + m15];
        const v2f vb2 = wp[32 + m15];
        const v2f vb3 = wp[48 + m15];

        acc0 = __builtin_amdgcn_wmma_f32_16x16x4_f32(false, a, false, vb0, (short)0, acc0, false, false);
        acc1 = __builtin_amdgcn_wmma_f32_16x16x4_f32(false, a, false, vb1, (short)0, acc1, false, false);
        acc2 = __builtin_amdgcn_wmma_f32_16x16x4_f32(false, a, false, vb2, (short)0, acc2, false, false);
        acc3 = __builtin_amdgcn_wmma_f32_16x16x4_f32(false, a, false, vb3, (short)0, acc3, false, false);
    }

    // C/D layout: VGPR r -> M = r (lanes 0-15) or r+8 (lanes 16-31), N = lane&15
    const int   mlo = (lane >> 4) * 8;
    const float bz0 = b1[ 0 + m15], bz1 = b1[16 + m15],
                bz2 = b1[32 + m15], bz3 = b1[48 + m15];
#pragma unroll
    for (int r = 0; r < 8; ++r) {
        const int m = r + mlo;
        hid[wave][m * HIDDEN +  0 + m15] = fmaxf(acc0[r] + bz0, 0.0f);
        hid[wave][m * HIDDEN + 16 + m15] = fmaxf(acc1[r] + bz1, 0.0f);
        hid[wave][m * HIDDEN + 32 + m15] = fmaxf(acc2[r] + bz2, 0.0f);
        hid[wave][m * HIDDEN + 48 + m15] = fmaxf(acc3[r] + bz3, 0.0f);
    }
    __syncthreads();                                       // uniform barrier

    // tiny second GEMM (64 -> 4): each lane computes 2 of the 64 tile outputs
    const int m     = lane & 15;
    const int jbase = (lane >> 4) * 2;    // lanes 0-15: j=0,1 ; lanes 16-31: j=2,3
    float s0 = b2[jbase], s1 = b2[jbase + 1];
    for (int n = 0; n < HIDDEN; ++n) {
        const float h = hid[wave][m * HIDDEN + n];
        s0 = fmaf(h, W2[n * N_FILT + jbase    ], s0);
        s1 = fmaf(h, W2[n * N_FILT + jbase + 1], s1);
    }
    if (valid) {
        F[(size_t)(row0 + m) * N_FILT + jbase    ] = s0;
        F[(size_t)(row0 + m) * N_FILT + jbase + 1] = s1;
    }
}

// ---------------------------------------------------------------------------
// CSR adjacency of directed edges grouped by u (filtration-independent).
// ---------------------------------------------------------------------------
__global__ void togl_zero_kernel(int* __restrict__ deg)
{
    const int i = blockIdx.x * blockDim.x + threadIdx.x;
    if (i < N_NODES) deg[i] = 0;
}

__global__ void togl_hist_kernel(const int* __restrict__ el, int* __restrict__ deg)
{
    const int i = blockIdx.x * blockDim.x + threadIdx.x;
    if (i < N_EDGES) atomicAdd(&deg[el[i]], 1);
}

__global__ __launch_bounds__(1024)
void togl_scan_kernel(const int* __restrict__ deg, int* __restrict__ row_off,
                      int* __restrict__ cursor)
{
    __shared__ int part[1024];
    const int tid  = threadIdx.x;
    const int CH   = (N_NODES + 1023) / 1024;    // 10 nodes per thread
    const int base = tid * CH;

    int s = 0;
    for (int i = 0; i < CH; ++i) {
        const int idx = base + i;
        if (idx < N_NODES) s += deg[idx];
    }
    part[tid] = s;
    __syncthreads();
    for (int off = 1; off < 1024; off <<= 1) {
        int v = 0;
        if (tid >= off) v = part[tid - off];
        __syncthreads();
        part[tid] += v;
        __syncthreads();
    }
    int run = (tid == 0) ? 0 : part[tid - 1];
    for (int i = 0; i < CH; ++i) {
        const int idx = base + i;
        if (idx < N_NODES) {
            row_off[idx] = run;
            cursor[idx]  = run;
            run += deg[idx];
        }
    }
    if (tid == 1023) row_off[N_NODES] = part[1023];
}

__global__ void togl_scatter_kernel(const int* __restrict__ el, int* __restrict__ cursor,
                                    int* __restrict__ adj)
{
    const int i = blockIdx.x * blockDim.x + threadIdx.x;
    if (i < N_EDGES) {
        const int u = el[i];
        const int v = el[N_EDGES + i];
        const int p = atomicAdd(&cursor[u], 1);
        adj[p] = v;
    }
}

// ---------------------------------------------------------------------------
// Per-filtration persistence. One workgroup per filtration.
// Stable rank via 64-bit keys (order-preserving f32 bits || index), LDS
// bitonic sort (128 KB), then single-lane union-find entirely in LDS.
// Deaths written straight to d_out (pre-set to +inf == "never merged").
// ---------------------------------------------------------------------------
__global__ __launch_bounds__(1024)
void togl_pers_kernel(const float* __restrict__ F, const int* __restrict__ row_off,
                      const int* __restrict__ adj, float* __restrict__ out)
{
    extern __shared__ unsigned int lds[];
    unsigned long long* keys   = (unsigned long long*)lds;        // 16384 * 8 B
    int*                order  = (int*)(lds + SORT_N * 2);        // 10000
    int*                pos    = (int*)(lds + SORT_N * 2 + N_NODES);
    int*                parent = (int*)(lds + SORT_N * 2 + 2 * N_NODES);

    const int f   = blockIdx.x;
    const int tid = threadIdx.x;
    const int NT  = blockDim.x;
    float* of = out + (size_t)f * N_NODES * 2;

    // build sortable keys: monotone uint mapping of f32, index in low bits (stable ties)
    for (int i = tid; i < SORT_N; i += NT) {
        unsigned long long kk;
        if (i < N_NODES) {
            const unsigned int bits = __float_as_uint(F[(size_t)i * N_FILT + f]);
            const unsigned int s = (bits & 0x80000000u) ? ~bits : (bits | 0x80000000u);
            kk = ((unsigned long long)s << 32) | (unsigned int)i;
        } else {
            kk = ~0ull;                                          // padding sinks to the end
        }
        keys[i] = kk;
    }
    __syncthreads();

    // bitonic sort in LDS
    for (int k = 2; k <= SORT_N; k <<= 1) {
        for (int j = k >> 1; j > 0; j >>= 1) {
            for (int idx = tid; idx < SORT_N; idx += NT) {
                const int ixj = idx ^ j;
                if (ixj > idx) {
                    const unsigned long long a = keys[idx];
                    const unsigned long long b = keys[ixj];
                    const bool up = (idx & k) == 0;
                    if ((a > b) == up) { keys[idx] = b; keys[ixj] = a; }
                }
            }
            __syncthreads();
        }
    }

    for (int t = tid; t < N_NODES; t += NT)
        order[t] = (int)(unsigned int)(keys[t] & 0xFFFFFFFFu);
    __syncthreads();
    for (int t = tid; t < N_NODES; t += NT) pos[order[t]] = t;
    for (int i = tid; i < N_NODES; i += NT) parent[i] = i;
    for (int i = tid; i < N_NODES; i += NT) {
        of[i * 2 + 0] = (float)i;                 // births = arange(n)
        of[i * 2 + 1] = __builtin_inff();         // deaths default = inf
    }
    __syncthreads();

    // sequential elder-rule union-find (LDS-resident), nodes in rank order.
    // An edge (u,v) fires exactly once, at t = max(pos[u],pos[v]) — the
    // reversed duplicate is the reference's self-loop no-op.
    if (tid == 0) {
        for (int t = 1; t < N_NODES; ++t) {
            const int x  = order[t];
            const int e0 = row_off[x];
            const int e1 = row_off[x + 1];
            if (t + 1 < N_NODES)                      // hide latency of next adjacency row
                __builtin_prefetch(&adj[row_off[order[t + 1]]], 0, 1);
            for (int e = e0; e < e1; ++e) {
                const int v = adj[e];
                if (pos[v] < t) {
                    int rx = x;
                    while (parent[rx] != rx) { parent[rx] = parent[parent[rx]]; rx = parent[rx]; }
                    int rv = v;
                    while (parent[rv] != rv) { parent[rv] = parent[parent[rv]]; rv = parent[rv]; }
                    if (rx != rv) {
                        const int px      = pos[rx];
                        const int pv      = pos[rv];
                        const int older   = (px <= pv) ? rx : rv;
                        const int younger = rx + rv - older;
                        parent[younger] = older;
                        of[(size_t)pos[younger] * 2 + 1] = (float)t;   // death = rank time
                    }
                }
            }
        }
    }
}

// ---------------------------------------------------------------------------
extern "C" void kernel_launch(void* const* d_in, const int* in_sizes, int n_in,
                              void* d_out, int out_size, void* d_ws, size_t ws_size,
                              hipStream_t stream)
{
    const float* X  = (const float*)d_in[0];
    const int*   el = (const int*)  d_in[1];   // (2, 80000) row-major int32
    const float* W1 = (const float*)d_in[2];
    const float* b1 = (const float*)d_in[3];
    const float* W2 = (const float*)d_in[4];
    const float* b2 = (const float*)d_in[5];
    float* out = (float*)d_out;                // (4, 10000, 2) f32

    // workspace layout (int granularity)
    int*   wsi     = (int*)d_ws;
    float* F       = (float*)d_ws;             // 40000 f32
    int*   deg     = wsi + 40000;              // 10000
    int*   row_off = wsi + 50000;              // 10001
    int*   cursor  = wsi + 60004;              // 10000
    int*   adj     = wsi + 70004;              // 80000  (total ~600 KB)

    // allow 251 KB dynamic LDS (CDNA5 WGP has 320 KB)
    (void)hipFuncSetAttribute(reinterpret_cast<const void*>(togl_pers_kernel),
                              hipFuncAttributeMaxDynamicSharedMemorySize,
                              PERS_LDS_BYTES);

    const int n_gemm_blocks = (N_TILES + WAVES_PB - 1) / WAVES_PB;   // 157

    togl_zero_kernel   <<<(N_NODES + 255) / 256, 256, 0, stream>>>(deg);
    togl_gemm_kernel   <<<n_gemm_blocks, WAVES_PB * 32, 0, stream>>>(X, W1, b1, W2, b2, F);
    togl_hist_kernel   <<<(N_EDGES + 255) / 256, 256, 0, stream>>>(el, deg);
    togl_scan_kernel   <<<1, 1024, 0, stream>>>(deg, row_off, cursor);
    togl_scatter_kernel<<<(N_EDGES + 255) / 256, 256, 0, stream>>>(el, cursor, adj);
    togl_pers_kernel   <<<N_FILT, 1024, PERS_LDS_BYTES, stream>>>(F, row_off, adj, out);
}